// SparseCoder_3023656976561
// MI455X (gfx1250) — compile-verified
//
#include <hip/hip_runtime.h>
#include <math.h>

// ---------------------------------------------------------------------------
// SparseCoder for MI455X (gfx1250, wave32).
//
// Algebraic collapse: per graph
//   P  = V @ E                       [256 x 8]     (WMMA f32 16x16x4)
//   S  = P^T P                       [8 x 8]
//   wt = (P^T x).flatten (c-major)   [256]
//   U[c*8+k, a] = A[k, c*256+a]      (pure index remap of A)
//   W  = blockdiag(S) * U            [256 x 256]   (built in 32-row chunks)
//   G  = U^T W                       [256 x 256]   (WMMA f32 16x16x4, in LDS)
//   b  = U^T wt                      [256]
// then 15 power iterations (Lipschitz L) + 50 FISTA iterations using only
// LDS-resident G, b, masks. One workgroup (256 thr = 8 waves) per graph.
//
// E is padded to 16 columns (cols 8..15 = 0) so the WMMA B-fragment loads in
// the hot loop are unconditional ds_load_b32 (no EXEC predication).
//
// LDS budget: G 256KB + ~48KB scratch = 311,552 B < 320KB/WGP.
// ---------------------------------------------------------------------------

typedef float v2f __attribute__((ext_vector_type(2)));
typedef float v8f __attribute__((ext_vector_type(8)));

#define NB      32
#define NN      256
#define NC      32
#define NK      8
#define NKP     16                   // padded E column stride
#define NATOMS  256
#define F_LAM   0.1f
#define NFISTA  50
#define NPOWER  15
#define ATOLE   1e-6f

#define LDS_FLOATS (65536 + 12352)   // G + scratch

extern "C" __global__ void __launch_bounds__(256, 1)
sparse_coder_kernel(const float* __restrict__ x,
                    const float* __restrict__ V,
                    const float* __restrict__ eigs,
                    const int*   __restrict__ y,
                    const float* __restrict__ A,
                    float*       __restrict__ out)
{
    extern __shared__ float sm[];
    float* G    = sm;             // 65536 floats, G[j*256+a]
    float* scr  = sm + 65536;
    float* E    = scr;            // 4096  (phase A: E[n*16+k], cols 8..15 zero)
    float* P    = scr + 4096;     // 2048  (phase A: P[n*8+k])
    float* Wc   = scr;            // 8192  (phase G: 32x256 chunk; aliases E/P)
    float* S    = scr + 8192;     // 64
    float* wt   = scr + 8256;     // 256
    float* bv   = scr + 8512;     // 256
    float* zv   = scr + 8768;     // 256
    float* rv   = scr + 9024;     // 256
    float* minv = scr + 9280;     // 256
    float* mexv = scr + 9536;     // 256
    float* red  = scr + 9792;     // 256
    float* pv   = scr + 10048;    // 256

    const int b    = blockIdx.x;
    const int tid  = threadIdx.x;     // 0..255
    const int lane = tid & 31;
    const int wv   = tid >> 5;        // wave 0..7
    const int lo   = lane & 15;       // lane % 16
    const int hi   = lane >> 4;       // 0/1 half of wave32

    const float* Vg = V + (size_t)b * NN * NN;
    const float* xg = x + (size_t)b * NN * NC;
    const float* Ag = A;              // [8][8192]

    // warm L2 for this graph's V slab (global_prefetch_b8)
    __builtin_prefetch(Vg + (size_t)tid * NN, 0, 3);

    // ---- Phase 1: spectral monomials E[n][k] = eig_n^k (atol mask, padded) -
    {
        float e = eigs[b * NN + tid];
        if (fabsf(e) < ATOLE) e = 0.f;
        float p = 1.f;
        #pragma unroll
        for (int k = 0; k < NK; ++k) { E[tid * NKP + k] = p; p *= e; }
        #pragma unroll
        for (int k = NK; k < NKP; ++k) E[tid * NKP + k] = 0.f;
    }
    __syncthreads();

    // ---- Phase 2: P = V @ E via v_wmma_f32_16x16x4_f32 -------------------
    // 16 row tiles of 16; wave wv owns tiles 2*wv, 2*wv+1. Only cols 0..7 valid.
    for (int it = 0; it < 2; ++it) {
        const int i = wv * 2 + it;
        v8f acc = {0.f, 0.f, 0.f, 0.f, 0.f, 0.f, 0.f, 0.f};
        for (int j0 = 0; j0 < NN; j0 += 4) {
            const int kk = hi * 2;
            v2f af, bf;
            // A-frag (16x4 f32): lane lo holds rows, VGPR0/1 = K = 2*hi, 2*hi+1
            af.x = Vg[(i * 16 + lo) * NN + j0 + kk];
            af.y = Vg[(i * 16 + lo) * NN + j0 + kk + 1];
            // B-frag (4x16 f32): row striped across lanes; padding is data
            bf.x = E[(j0 + kk) * NKP + lo];
            bf.y = E[(j0 + kk + 1) * NKP + lo];
            acc = __builtin_amdgcn_wmma_f32_16x16x4_f32(
                false, af, false, bf, (short)0, acc, false, false);
        }
        if (lo < NK) {
            #pragma unroll
            for (int v = 0; v < 8; ++v)               // C/D: vgpr v -> M = v (+8 for hi)
                P[(i * 16 + v + hi * 8) * NK + lo] = acc[v];
        }
    }
    __syncthreads();

    // ---- Phase 3: S = P^T P (8x8) ----------------------------------------
    if (tid < 64) {
        const int k = tid >> 3, k2 = tid & 7;
        float a1 = 0.f;
        for (int n = 0; n < NN; ++n) a1 += P[n * NK + k] * P[n * NK + k2];
        S[tid] = a1;
    }
    // ---- Phase 4: wt[c*8+k] = sum_n P[n][k] * x[n][c] --------------------
    {
        const int c = tid >> 3, k = tid & 7;
        float a1 = 0.f;
        for (int n = 0; n < NN; ++n) a1 += P[n * NK + k] * xg[n * NC + c];
        wt[tid] = a1;
    }
    __syncthreads();

    // ---- Phase 5: G = U^T W, chunked over 32 rows of ck ------------------
    for (int q = tid; q < 65536; q += 256) G[q] = 0.f;
    __syncthreads();

    for (int r0 = 0; r0 < 256; r0 += 32) {
        // build Wc[rr][a] = sum_k' S[k][k'] * U[c*8+k'][a], rows ck = r0+rr
        {
            const int cbase = r0 >> 3;                // 4 channels per chunk
            for (int cc = 0; cc < 4; ++cc) {
                const int c = cbase + cc;
                float av[8];
                #pragma unroll
                for (int k2 = 0; k2 < 8; ++k2)
                    av[k2] = Ag[k2 * 8192 + c * 256 + tid];
                #pragma unroll
                for (int k = 0; k < 8; ++k) {
                    float a1 = 0.f;
                    #pragma unroll
                    for (int k2 = 0; k2 < 8; ++k2) a1 += S[k * 8 + k2] * av[k2];
                    Wc[(cc * 8 + k) * 256 + tid] = a1;
                }
            }
        }
        __syncthreads();

        // G tiles: 16x16 grid of 16x16 tiles; wave owns 32 tiles; 8 K-steps/chunk
        for (int t = 0; t < 32; ++t) {
            const int ti = wv * 32 + t;
            const int i = ti >> 4, j = ti & 15;
            const int col = j * 16 + lo;
            v8f acc;
            #pragma unroll
            for (int v = 0; v < 8; ++v)
                acc[v] = G[(i * 16 + v + hi * 8) * 256 + col];
            #pragma unroll
            for (int ks = 0; ks < 8; ++ks) {
                const int kk  = ks * 4 + hi * 2;      // local row in chunk
                const int ck0 = r0 + kk;
                const int ck1 = ck0 + 1;
                const int a0  = i * 16 + lo;
                v2f af, bf;
                // A-frag = U[ck, a]  (U is an index remap of A)
                af.x = Ag[(ck0 & 7) * 8192 + (ck0 >> 3) * 256 + a0];
                af.y = Ag[(ck1 & 7) * 8192 + (ck1 >> 3) * 256 + a0];
                // B-frag = W[ck, a'] from LDS chunk
                bf.x = Wc[kk * 256 + col];
                bf.y = Wc[(kk + 1) * 256 + col];
                acc = __builtin_amdgcn_wmma_f32_16x16x4_f32(
                    false, af, false, bf, (short)0, acc, false, false);
            }
            #pragma unroll
            for (int v = 0; v < 8; ++v)
                G[(i * 16 + v + hi * 8) * 256 + col] = acc[v];
        }
        __syncthreads();
    }

    // ---- Phase 6: b = U^T wt ---------------------------------------------
    {
        float a1 = 0.f;
        for (int c = 0; c < NC; ++c)
            #pragma unroll
            for (int k = 0; k < 8; ++k)
                a1 += Ag[k * 8192 + c * 256 + tid] * wt[c * 8 + k];
        bv[tid] = a1;
    }

    // ---- Phase 7: masks from label ---------------------------------------
    {
        const int yi = y[b];                // 0..5 -> PARTITION[i] = 32*i
        const int start = 32 * yi, end = start + 32;
        const bool lab = (tid >= start) && (tid < end);
        minv[tid] = (lab || tid >= 192) ? 1.f : 0.f;
        mexv[tid] = ((tid < start) || (tid >= end && tid < 192)) ? 1.f : 0.f;
        pv[tid]   = 1.f / 16.f;             // 1/sqrt(256)
    }
    __syncthreads();

    // ---- Phase 8: power iteration for Lipschitz L ------------------------
    // gmul(v) = G v + m_in .* (G (m_in .* v)) + m_ex .* v   (G symmetric)
    float L = 0.f;
    for (int it = 0; it < NPOWER; ++it) {
        float a1 = 0.f, a2 = 0.f;
        for (int jx = 0; jx < 256; ++jx) {
            const float g  = G[jx * 256 + tid];     // conflict-free: lane-adjacent
            const float vj = pv[jx];                // broadcast
            a1 += g * vj;
            a2 += g * (minv[jx] * vj);
        }
        const float w = a1 + minv[tid] * a2 + mexv[tid] * pv[tid];
        red[tid] = w * w;
        __syncthreads();
        for (int s = 128; s > 0; s >>= 1) {
            if (tid < s) red[tid] += red[tid + s];
            __syncthreads();
        }
        const float nw = sqrtf(red[0]);
        pv[tid] = w / (nw + 1e-12f);
        L = nw;
        __syncthreads();
    }
    L += 1e-6f;

    // ---- Phase 9: FISTA ---------------------------------------------------
    zv[tid] = 0.f;
    rv[tid] = 0.f;
    __syncthreads();
    float tcur = 1.f;
    const float invL = 1.f / L;
    const float thr  = F_LAM / L;
    for (int it = 0; it < NFISTA; ++it) {
        float a1 = 0.f, a2 = 0.f;
        for (int jx = 0; jx < 256; ++jx) {
            const float g  = G[jx * 256 + tid];
            const float zj = zv[jx];
            a1 += g * zj;
            a2 += g * (minv[jx] * zj);
        }
        const float grad = (a1 - bv[tid]) + minv[tid] * (a2 - bv[tid])
                         + mexv[tid] * zv[tid];
        const float u  = zv[tid] - grad * invL;
        const float rn = copysignf(fmaxf(fabsf(u) - thr, 0.f), u);
        const float tn = 0.5f * (1.f + sqrtf(1.f + 4.f * tcur * tcur));
        __syncthreads();                      // all reads of zv done
        const float rold = rv[tid];
        zv[tid] = rn + ((tcur - 1.f) / tn) * (rn - rold);
        rv[tid] = rn;
        tcur = tn;
        __syncthreads();
    }

    out[b * NATOMS + tid] = rv[tid];          // output shape [B, ATOMS, 1]
}

extern "C" void kernel_launch(void* const* d_in, const int* in_sizes, int n_in,
                              void* d_out, int out_size, void* d_ws, size_t ws_size,
                              hipStream_t stream) {
    (void)in_sizes; (void)n_in; (void)out_size; (void)d_ws; (void)ws_size;
    const float* x    = (const float*)d_in[0];
    const float* V    = (const float*)d_in[1];
    const float* eigs = (const float*)d_in[2];
    const int*   y    = (const int*)d_in[3];
    const float* A    = (const float*)d_in[4];
    float* out = (float*)d_out;

    const size_t lds = (size_t)LDS_FLOATS * sizeof(float);   // 311,552 B < 320KB
    hipFuncSetAttribute(reinterpret_cast<const void*>(sparse_coder_kernel),
                        hipFuncAttributeMaxDynamicSharedMemorySize, (int)lds);
    sparse_coder_kernel<<<dim3(NB), dim3(256), lds, stream>>>(x, V, eigs, y, A, out);
}